// CfCDoseController_27384711479693
// MI455X (gfx1250) — compile-verified
//
#include <hip/hip_runtime.h>

// CfC (liquid NN) recurrence on gfx1250 via V_WMMA_F32_16X16X4_F32.
// One wave32 handles 16 batch sequences; M = gate rows, N = batch, K = features.
// K-order permuted so WMMA B operands are raw D-tile register pairs (no shuffles).

typedef __attribute__((ext_vector_type(2))) float v2f;
typedef __attribute__((ext_vector_type(8))) float v8f;

#define B_SZ 4096
#define T_SZ 512
#define SIGMA_MIN_F 0.001f

struct Params {
  const float* x;        // [B, T, 4]
  const float* w[27];    // per layer: ff1_w, ff1_b, ff2_w, ff2_b, ta_w, ta_b, tb_w, tb_b, mask
  const float* sscale;   // [1]
  float* sigma;          // [B, T]
  float* hx;             // [B, 16]
};

__device__ __forceinline__ float ftanh(float x) {
#if __has_builtin(__builtin_amdgcn_tanhf)
  return __builtin_amdgcn_tanhf(x);   // v_tanh_f32 (gfx1250 trans op)
#else
  return tanhf(x);
#endif
}

// Which gate-output index j does tile output-row m hold for layer l? (-1 = unused row)
__device__ __forceinline__ int outrow_to_j(int l, int m) {
  if (l == 0) return (m < 9) ? m : -1;              // h0 feats at rows 0..8
  if (l == 1) return (m >= 9 && m < 15) ? m - 9 : -1; // h1 feats at rows 9..14
  return (m == 0) ? 0 : -1;                          // h2 feat at row 0
}

// Which weight column k (of the xx=[inp,h] concat) does tile feature-row r hold?
__device__ __forceinline__ int tilerow_to_k(int l, int r) {
  if (l == 0) {                      // xx0 = [x_t(4) @ cols 0..3, h0(9) @ cols 4..12]
    if (r < 9)  return 4 + r;        // h0 feats at rows 0..8
    if (r < 13) return r - 9;        // x_t feats at rows 9..12
    return -1;
  }
  if (l == 1) {                      // xx1 = [h0n(9) cols 0..8 rows 0..8, h1(6) cols 9..14 rows 9..14]
    return (r < 15) ? r : -1;
  }
  // l == 2: xx2 = [h1n(6) cols 0..5 rows 9..14, h2(1) col 6 row 0]
  if (r == 0) return 6;
  if (r >= 9 && r < 15) return r - 9;
  return -1;
}

// Three independent gate accumulations (ff1, ff2, ta+tb) sharing the same B operand.
// Permuted-K chunks: chunk c = feature rows {2c, 2c+1, 2c+8, 2c+9}  ->  B = (xx[2c], xx[2c+1]).
__device__ __forceinline__ void gates3(const float (&A)[3][4][2],
                                       const float (&bz)[3][8],
                                       const float (&xx)[8],
                                       v8f& g0, v8f& g1, v8f& g2) {
#pragma unroll
  for (int q = 0; q < 8; q++) { g0[q] = bz[0][q]; g1[q] = bz[1][q]; g2[q] = bz[2][q]; }
#pragma unroll
  for (int c = 0; c < 4; c++) {
    v2f b  = {xx[2 * c], xx[2 * c + 1]};
    v2f a0 = {A[0][c][0], A[0][c][1]};
    v2f a1 = {A[1][c][0], A[1][c][1]};
    v2f a2 = {A[2][c][0], A[2][c][1]};
    g0 = __builtin_amdgcn_wmma_f32_16x16x4_f32(false, a0, false, b, (short)0, g0, false, false);
    g1 = __builtin_amdgcn_wmma_f32_16x16x4_f32(false, a1, false, b, (short)0, g1, false, false);
    g2 = __builtin_amdgcn_wmma_f32_16x16x4_f32(false, a2, false, b, (short)0, g2, false, false);
  }
}

__global__ __launch_bounds__(32) void cfc_wmma_kernel(Params p) {
  const int  lane = threadIdx.x & 31;
  const int  m    = lane & 15;     // tile row index my lane contributes to (A:M, B/D: column m = batch)
  const bool hi   = lane >= 16;    // hi half holds rows 8..15 of D tiles / K 2..3 of A,B chunks
  const int  b0   = blockIdx.x * 16;
  const int  bcol = b0 + m;        // my batch column

  // ---- Build loop-invariant A-operands (permuted-K layout) and bias C-tiles in registers ----
  float wA[3][3][4][2];   // [layer][gate][chunk][kv]
  float bC[3][3][8];      // bias tiles: vq = bias(row q | q+8)
#pragma unroll
  for (int l = 0; l < 3; l++) {
    const int catl = (l == 0) ? 13 : ((l == 1) ? 15 : 7);
    const float* W1 = p.w[l * 9 + 0]; const float* B1 = p.w[l * 9 + 1];
    const float* W2 = p.w[l * 9 + 2]; const float* B2 = p.w[l * 9 + 3];
    const float* WA = p.w[l * 9 + 4]; const float* BA = p.w[l * 9 + 5];
    const float* WB = p.w[l * 9 + 6]; const float* BB = p.w[l * 9 + 7];
    const float* MK = p.w[l * 9 + 8];
    const int j = outrow_to_j(l, m);
#pragma unroll
    for (int c = 0; c < 4; c++) {
#pragma unroll
      for (int v = 0; v < 2; v++) {
        const int r = 2 * c + v + (hi ? 8 : 0);   // feature tile-row this (lane, c, v) supplies
        const int k = tilerow_to_k(l, r);
        float w0 = 0.f, w1 = 0.f, w2 = 0.f;
        if (j >= 0 && k >= 0) {
          const int idx = j * catl + k;
          const float mk = MK[idx];
          w0 = W1[idx] * mk;          // ff1 (sparsity-masked)
          w1 = W2[idx] * mk;          // ff2 (sparsity-masked)
          w2 = WA[idx] + WB[idx];     // fused time gate (dense)
        }
        wA[l][0][c][v] = w0; wA[l][1][c][v] = w1; wA[l][2][c][v] = w2;
      }
    }
#pragma unroll
    for (int q = 0; q < 8; q++) {
      const int r  = q + (hi ? 8 : 0);
      const int jb = outrow_to_j(l, r);
      bC[l][0][q] = (jb >= 0) ? B1[jb] : 0.f;
      bC[l][1][q] = (jb >= 0) ? B2[jb] : 0.f;
      bC[l][2][q] = (jb >= 0) ? (BA[jb] + BB[jb]) : 0.f;
    }
  }

  const float sscale = p.sscale[0];

  // ---- Recurrent state tiles (unused rows are exactly 0 by construction) ----
  float h0[8], h1[8];
#pragma unroll
  for (int q = 0; q < 8; q++) { h0[q] = 0.f; h1[q] = 0.f; }
  float h2 = 0.f;

  const float* xrow = p.x + (size_t)bcol * T_SZ * 4;
  float4 xc = *(const float4*)(xrow);  // x_t prefetch pipeline

  for (int t = 0; t < T_SZ; t += 4) {
    float s4[4];
#pragma unroll
    for (int k4 = 0; k4 < 4; k4++) {
      const int t_ = t + k4;
      const int tn = (t_ + 1 < T_SZ) ? t_ + 1 : t_;
      const float4 xn = *(const float4*)(xrow + (size_t)tn * 4);  // prefetch next step

      // ----- layer 0: xx0 = h0 rows 0..8, x_t rows 9..12 -----
      float xx[8];
      xx[0] = h0[0];
      xx[1] = hi ? xc.x : h0[1];
      xx[2] = hi ? xc.y : h0[2];
      xx[3] = hi ? xc.z : h0[3];
      xx[4] = hi ? xc.w : h0[4];
      xx[5] = h0[5]; xx[6] = h0[6]; xx[7] = h0[7];
      v8f g0, g1, g2;
      gates3(wA[0], bC[0], xx, g0, g1, g2);
      float h0n[8];
#pragma unroll
      for (int q = 0; q < 8; q++) {
        const float f1 = ftanh(g0[q]);
        const float f2 = ftanh(g1[q]);
        const float ti = 0.5f * ftanh(0.5f * g2[q]) + 0.5f;   // sigmoid
        h0n[q] = f1 + ti * (f2 - f1);
      }

      // ----- layer 1: xx1 = h0n rows 0..8, h1 rows 9..14 -----
      float xx1[8];
      xx1[0] = h0n[0];
#pragma unroll
      for (int q = 1; q < 7; q++) xx1[q] = hi ? h1[q] : h0n[q];
      xx1[7] = h0n[7];
      gates3(wA[1], bC[1], xx1, g0, g1, g2);
      float h1n[8];
#pragma unroll
      for (int q = 0; q < 8; q++) {
        const float f1 = ftanh(g0[q]);
        const float f2 = ftanh(g1[q]);
        const float ti = 0.5f * ftanh(0.5f * g2[q]) + 0.5f;
        h1n[q] = f1 + ti * (f2 - f1);
      }

      // ----- layer 2: xx2 = h1n rows 9..14, h2 row 0 -----
      float xx2[8];
      xx2[0] = hi ? h1n[0] : h2;
#pragma unroll
      for (int q = 1; q < 8; q++) xx2[q] = h1n[q];
      gates3(wA[2], bC[2], xx2, g0, g1, g2);
      {
        const float f1 = ftanh(g0[0]);
        const float f2 = ftanh(g1[0]);
        const float ti = 0.5f * ftanh(0.5f * g2[0]) + 0.5f;
        h2 = f1 + ti * (f2 - f1);   // motor output == new h2
      }

#pragma unroll
      for (int q = 0; q < 8; q++) { h0[q] = h0n[q]; h1[q] = h1n[q]; }

      const float sg = 0.5f * ftanh(0.5f * h2) + 0.5f;        // sigmoid(out)
      s4[k4] = sg * sscale + SIGMA_MIN_F;
      xc = xn;
    }
    if (!hi) {  // rows 0..15 of sigma tile live in lanes 0..15 (row 0 = motor)
      float4 o; o.x = s4[0]; o.y = s4[1]; o.z = s4[2]; o.w = s4[3];
      *(float4*)(p.sigma + (size_t)bcol * T_SZ + t) = o;      // t-contiguous b128 store
    }
  }

  // ---- final hidden concat hx = [h0(9), h1(6), h2(1)] ----
  const size_t hb = (size_t)bcol * 16;
  if (!hi) {
#pragma unroll
    for (int q = 0; q < 8; q++) p.hx[hb + q] = h0[q];   // h0 feats 0..7 (lo rows)
    p.hx[hb + 15] = h2;                                 // h2 feat
  } else {
    p.hx[hb + 8] = h0[0];                               // h0 feat 8 (row 8 = hi half of v0)
#pragma unroll
    for (int q = 1; q < 7; q++) p.hx[hb + 9 + (q - 1)] = h1[q];  // h1 feats 0..5 (rows 9..14)
  }
}

extern "C" void kernel_launch(void* const* d_in, const int* in_sizes, int n_in,
                              void* d_out, int out_size, void* d_ws, size_t ws_size,
                              hipStream_t stream) {
  (void)in_sizes; (void)n_in; (void)out_size; (void)d_ws; (void)ws_size;
  Params p;
  p.x = (const float*)d_in[0];
  for (int i = 0; i < 27; i++) p.w[i] = (const float*)d_in[1 + i];
  p.sscale = (const float*)d_in[28];
  p.sigma  = (float*)d_out;
  p.hx     = (float*)d_out + (size_t)B_SZ * T_SZ;
  hipLaunchKernelGGL(cfc_wmma_kernel, dim3(B_SZ / 16), dim3(32), 0, stream, p);
}